// WaveletFeatureExtractor_62053687492849
// MI455X (gfx1250) — compile-verified
//
#include <hip/hip_runtime.h>

// ---------------------------------------------------------------------------
// Haar 4-level DWT over rows of (M x 64) fp32, cast as out = A * W^T using
// CDNA5 v_wmma_f32_16x16x4_f32. Memory-bound: 1.07 GB traffic -> ~46us at
// 23.3 TB/s; the 17 GFLOP dense fp32 GEMM is free on the matrix pipes.
//
// B fragments are pre-paired as float2 in LDS (one ds_load_b64 per fragment,
// no register gluing) with a lane-half rotation that makes all B reads
// bank-conflict-free and lets every k-chunk use immediate DS offsets.
// A loads / D stores are nontemporal (pure streaming, no L2 reuse).
// ---------------------------------------------------------------------------

typedef __attribute__((ext_vector_type(2))) float v2f;
typedef __attribute__((ext_vector_type(8))) float v8f;

#define WAVES_PER_BLOCK  8
#define TILES_PER_WAVE   4
#define ROWS_PER_BLOCK   (WAVES_PER_BLOCK * 16 * TILES_PER_WAVE)   // 512

// W[n][j]: row n of the 64x64 Haar analysis matrix, pywt ordering
// [cA4 (4), cD4 (4), cD3 (8), cD2 (16), cD1 (32)].
__device__ __forceinline__ float haar_w(int n, int j) {
    if (n < 4) {                       // cA4: + (1/sqrt2)^4 over 16-blocks
        return ((j >> 4) == n) ? 0.25f : 0.0f;
    }
    int L, base;
    if      (n <  8) { L = 4; base =  4; }   // cD4
    else if (n < 16) { L = 3; base =  8; }   // cD3
    else if (n < 32) { L = 2; base = 16; }   // cD2
    else             { L = 1; base = 32; }   // cD1
    int k   = n - base;
    int bs  = 1 << L;
    int rel = j - k * bs;
    if (rel < 0 || rel >= bs) return 0.0f;
    float coef = (L == 1) ? 0.70710678118654752440f :
                 (L == 2) ? 0.5f :
                 (L == 3) ? 0.35355339059327376220f : 0.25f;
    return (rel < (bs >> 1)) ? coef : -coef;
}

__global__ __launch_bounds__(256)
void haar_wmma_kernel(const float* __restrict__ in, float* __restrict__ out, int M) {
    // Bm2[kc*128 + half*64 + npos] = { W^T[4kc+2*half + 0][n], W^T[...+1][n] }
    // with storage position npos = (n + half*16) & 63 (lane-half rotation).
    __shared__ v2f Bm2[16 * 2 * 64];                 // 16 KB

    const int tid = threadIdx.x;
    for (int i = tid; i < 16 * 2 * 64; i += 256) {
        const int kc    = i >> 7;
        const int half  = (i >> 6) & 1;
        const int npos  = i & 63;
        const int n     = (npos - half * 16) & 63;   // logical column
        const int kbase = kc * 4 + half * 2;
        v2f val;
        val.x = haar_w(n, kbase + 0);                // W^T[k][n] = W[n][k]
        val.y = haar_w(n, kbase + 1);
        Bm2[i] = val;
    }
    __syncthreads();

    const int lane = tid & 31;
    const int wave = tid >> 5;
    const int half = lane >> 4;        // 0: lanes 0-15, 1: lanes 16-31
    const int l16  = lane & 15;
    const int koff = half * 2;         // A-fragment K offset per ISA layout

    // Per-tile B fragment base pointers; k-chunk reached via immediate offset
    // (kc*128 v2f = kc*1024 bytes). Rotation is folded into the index so the
    // read at tile t is simply ((t*16 + lane) & 63) within the half's row.
    const int bbase = half * 64;
    const v2f* __restrict__ bp0 = &Bm2[bbase + ((lane +  0) & 63)];
    const v2f* __restrict__ bp1 = &Bm2[bbase + ((lane + 16) & 63)];
    const v2f* __restrict__ bp2 = &Bm2[bbase + ((lane + 32) & 63)];
    const v2f* __restrict__ bp3 = &Bm2[bbase + ((lane + 48) & 63)];

    const long blockBase = (long)blockIdx.x * ROWS_PER_BLOCK;

    for (int it = 0; it < TILES_PER_WAVE; ++it) {
        const long rowBase = blockBase + (long)(it * WAVES_PER_BLOCK + wave) * 16;
        if (rowBase + 16 > (long)M) break;          // wave-uniform; EXEC stays all-ones

        // A fragment source: lane holds row (rowBase + l16), K pair (kbase, kbase+1).
        const float* __restrict__ arow = in + (rowBase + (long)l16) * 64;

        v8f acc0 = {}; v8f acc1 = {}; v8f acc2 = {}; v8f acc3 = {};

#pragma unroll
        for (int kc = 0; kc < 16; ++kc) {
            const int kbase = kc * 4 + koff;
            // 8-byte aligned streaming float2 load -> global_load_b64 (NT).
            const v2f a = __builtin_nontemporal_load((const v2f*)(arow + kbase));

            const v2f b0 = bp0[kc * 128];            // ds_load_b64, imm offset
            const v2f b1 = bp1[kc * 128];
            const v2f b2 = bp2[kc * 128];
            const v2f b3 = bp3[kc * 128];

            acc0 = __builtin_amdgcn_wmma_f32_16x16x4_f32(false, a, false, b0,
                                                         (short)0, acc0, false, false);
            acc1 = __builtin_amdgcn_wmma_f32_16x16x4_f32(false, a, false, b1,
                                                         (short)0, acc1, false, false);
            acc2 = __builtin_amdgcn_wmma_f32_16x16x4_f32(false, a, false, b2,
                                                         (short)0, acc2, false, false);
            acc3 = __builtin_amdgcn_wmma_f32_16x16x4_f32(false, a, false, b3,
                                                         (short)0, acc3, false, false);
        }

        // D layout: VGPR v -> row rowBase + v + half*8, lane l16 -> column.
        float* __restrict__ orow = out + (rowBase + (long)half * 8) * 64 + l16;
#pragma unroll
        for (int v = 0; v < 8; ++v) {
            float* __restrict__ o = orow + (long)v * 64;
            __builtin_nontemporal_store(acc0[v], o +  0);
            __builtin_nontemporal_store(acc1[v], o + 16);
            __builtin_nontemporal_store(acc2[v], o + 32);
            __builtin_nontemporal_store(acc3[v], o + 48);
        }
    }
}

extern "C" void kernel_launch(void* const* d_in, const int* in_sizes, int n_in,
                              void* d_out, int out_size, void* d_ws, size_t ws_size,
                              hipStream_t stream) {
    (void)n_in; (void)d_ws; (void)ws_size; (void)out_size;
    const float* in  = (const float*)d_in[0];
    float*       out = (float*)d_out;
    const int M = in_sizes[0] / 64;                   // 2097152 rows
    const int blocks = (M + ROWS_PER_BLOCK - 1) / ROWS_PER_BLOCK;  // 4096
    haar_wmma_kernel<<<blocks, 256, 0, stream>>>(in, out, M);
}